// U_32186484916947
// MI455X (gfx1250) — compile-verified
//
#include <hip/hip_runtime.h>

// Single-qubit unitary on qubit 0 (MSB axis) of a 2^24-amplitude state, batch 4.
// HBM-bound: ~1.07 GB traffic -> ~46 us floor at 23.3 TB/s. WMMA is useless at
// AI ~ 1 flop/byte; instead we exercise the CDNA5 async data-movement path:
// GLOBAL_LOAD_ASYNC_TO_LDS_B128 double-buffered software pipeline (ASYNCcnt),
// per-thread private LDS slots (no barriers), direct NT b128 stores out.

typedef float v4f __attribute__((ext_vector_type(4)));
typedef int   v4i __attribute__((__vector_size__(4 * sizeof(int))));

#define N_QUBITS 24
#define DIM      (1u << N_QUBITS)     // 16,777,216
#define HALF     (DIM >> 1)           // 8,388,608 pairs
#define BATCH    4
#define TPB      256
#define ITERS    8
#define PAIRS_PER_BLOCK (TPB * ITERS) // 2048 -> 4096 blocks

// Builtin prototype (from hipcc diagnostic): first arg is v4i* in the global
// (device) address space; LDS side is the shared address space.
typedef __attribute__((address_space(1))) v4i GV4;
typedef __attribute__((address_space(3))) v4i LV4;

#define CFENCE() asm volatile("" ::: "memory")

__device__ __forceinline__ void async_copy_b128(const void* gsrc, void* ldst) {
#if __has_builtin(__builtin_amdgcn_global_load_async_to_lds_b128)
    __builtin_amdgcn_global_load_async_to_lds_b128((GV4*)gsrc, (LV4*)ldst, 0, 0);
#else
    unsigned lofs = (unsigned)(unsigned long long)(LV4*)ldst;
    asm volatile("global_load_async_to_lds_b128 %0, %1, off"
                 :: "v"(lofs), "v"(gsrc) : "memory");
#endif
}

__device__ __forceinline__ void wait_async_le4() {
#if __has_builtin(__builtin_amdgcn_s_wait_asynccnt)
    __builtin_amdgcn_s_wait_asynccnt(4);
#else
    asm volatile("s_wait_asynccnt 0x4" ::: "memory");
#endif
}

__device__ __forceinline__ void wait_async_0() {
#if __has_builtin(__builtin_amdgcn_s_wait_asynccnt)
    __builtin_amdgcn_s_wait_asynccnt(0);
#else
    asm volatile("s_wait_asynccnt 0x0" ::: "memory");
#endif
}

__device__ __forceinline__ void wait_ds_0() {
#if __has_builtin(__builtin_amdgcn_s_wait_dscnt)
    __builtin_amdgcn_s_wait_dscnt(0);
#else
    asm volatile("s_wait_dscnt 0x0" ::: "memory");
#endif
}

// U = RZ(omega) RY(theta) RZ(phi) = [[a, b], [-conj(b), conj(a)]]
//   a = cos(t/2) * exp(-i(phi+omega)/2),  b = -sin(t/2) * conj(exp(-i(phi-omega)/2))
__global__ void u1q_coeffs_kernel(const float* __restrict__ thetas,
                                  float* __restrict__ coef) {
    int b = (int)threadIdx.x;
    if (b >= BATCH) return;
    float phi   = thetas[0 * BATCH + b];
    float theta = thetas[1 * BATCH + b];
    float omega = thetas[2 * BATCH + b];
    float s, c;
    sincosf(0.5f * theta, &s, &c);
    float sp, cp, sm, cm;
    sincosf(-0.5f * (phi + omega), &sp, &cp);   // t_plus  = cp + i*sp
    sincosf(-0.5f * (phi - omega), &sm, &cm);   // t_minus = cm + i*sm
    coef[0 * BATCH + b] =  c * cp;   // a_re
    coef[1 * BATCH + b] =  c * sp;   // a_im
    coef[2 * BATCH + b] = -s * cm;   // b_re
    coef[3 * BATCH + b] =  s * sm;   // b_im
}

__global__ __launch_bounds__(TPB) void u1q_apply_kernel(
    const v4f* __restrict__ sr,     // state_real: DIM rows of float4 (batch)
    const v4f* __restrict__ si,     // state_imag
    const float* __restrict__ coef, // 16 floats: a_re[4] a_im[4] b_re[4] b_im[4]
    v4f* __restrict__ out)          // (2, DIM, BATCH): real plane then imag plane
{
    __shared__ v4f stage[2][4][TPB];   // 32 KB double buffer, per-thread slots
    const unsigned tid  = threadIdx.x;
    const unsigned base = blockIdx.x * (unsigned)PAIRS_PER_BLOCK + tid;

    const v4f* cv = (const v4f*)coef;  // uniform address
    const v4f ar = cv[0];
    const v4f ai = cv[1];
    const v4f br = cv[2];
    const v4f bi = cv[3];

    // Prologue: async-prefetch stages 0 and 1 (8 outstanding b128 copies).
    {
        const unsigned i0 = base;
        async_copy_b128(sr + i0,        &stage[0][0][tid]);
        async_copy_b128(si + i0,        &stage[0][1][tid]);
        async_copy_b128(sr + HALF + i0, &stage[0][2][tid]);
        async_copy_b128(si + HALF + i0, &stage[0][3][tid]);
        const unsigned i1 = base + TPB;
        async_copy_b128(sr + i1,        &stage[1][0][tid]);
        async_copy_b128(si + i1,        &stage[1][1][tid]);
        async_copy_b128(sr + HALF + i1, &stage[1][2][tid]);
        async_copy_b128(si + HALF + i1, &stage[1][3][tid]);
    }

    v4f* const out_re = out;        // real plane: DIM*BATCH floats = DIM v4f
    v4f* const out_im = out + DIM;  // imag plane

#pragma unroll
    for (int k = 0; k < ITERS; ++k) {
        // In-order ASYNCcnt: <=4 outstanding means stage k's 4 copies landed.
        if (k < ITERS - 1) wait_async_le4();
        else               wait_async_0();
        CFENCE();

        const int sb = k & 1;
        v4f r0 = stage[sb][0][tid];
        v4f m0 = stage[sb][1][tid];
        v4f r1 = stage[sb][2][tid];
        v4f m1 = stage[sb][3][tid];

        // DS pipe and async engine are mutually unordered: make sure our DS
        // reads fully completed before the async engine may overwrite the slot.
        wait_ds_0();
        CFENCE();

        if (k + 2 < ITERS) {        // refill this buffer for stage k+2
            const unsigned in = base + (unsigned)(k + 2) * TPB;
            async_copy_b128(sr + in,        &stage[sb][0][tid]);
            async_copy_b128(si + in,        &stage[sb][1][tid]);
            async_copy_b128(sr + HALF + in, &stage[sb][2][tid]);
            async_copy_b128(si + HALF + in, &stage[sb][3][tid]);
        }

        // out0 = a*z0 + b*z1 ; out1 = -conj(b)*z0 + conj(a)*z1  (per batch lane)
        v4f or0 = ar * r0 - ai * m0 + br * r1 - bi * m1;
        v4f oi0 = ar * m0 + ai * r0 + br * m1 + bi * r1;
        v4f or1 = ar * r1 + ai * m1 - br * r0 - bi * m0;
        v4f oi1 = ar * m1 - ai * r1 - br * m0 + bi * r0;

        const unsigned i = base + (unsigned)k * TPB;
        __builtin_nontemporal_store(or0, out_re + i);
        __builtin_nontemporal_store(or1, out_re + HALF + i);
        __builtin_nontemporal_store(oi0, out_im + i);
        __builtin_nontemporal_store(oi1, out_im + HALF + i);
    }
}

extern "C" void kernel_launch(void* const* d_in, const int* in_sizes, int n_in,
                              void* d_out, int out_size, void* d_ws, size_t ws_size,
                              hipStream_t stream) {
    const float* state_real = (const float*)d_in[0];   // DIM*BATCH f32
    const float* state_imag = (const float*)d_in[1];   // DIM*BATCH f32
    const float* thetas     = (const float*)d_in[2];   // 3*BATCH f32
    float* out  = (float*)d_out;                       // 2*DIM*BATCH f32
    float* coef = (float*)d_ws;                        // 16 f32 scratch

    u1q_coeffs_kernel<<<1, 32, 0, stream>>>(thetas, coef);

    dim3 grid(HALF / PAIRS_PER_BLOCK);  // 4096 blocks x 256 threads (8 waves)
    u1q_apply_kernel<<<grid, TPB, 0, stream>>>(
        (const v4f*)state_real, (const v4f*)state_imag, coef, (v4f*)out);
}